// MFModel_36524401885311
// MI455X (gfx1250) — compile-verified
//
#include <hip/hip_runtime.h>
#include <hip/hip_bf16.h>

// ---------------------------------------------------------------------------
// MF loss for MI455X (gfx1250, wave32, WMMA).
//
// loss*n = [ sum(X^2) - 2*<W, X@H> + <W W^T, H^T H> ]          (recon)
//        + 0.5 * [ sum_k colsum_k^2 - sum(H^2) ]               (trace term)
//        + 0.01 * [ sum(H^2) + sum(W^2) ]                      (L2 terms)
//
// Big pass: Y[f,k] = sum_i X[f,i] * Hc[i,k] via V_WMMA_F32_16X16X4_F32,
// fused with sum(X^2) and the <W, Y> contraction. X is read exactly once.
// ---------------------------------------------------------------------------

#define NFEAT   4096
#define NBATCH  8192
#define NCLS    10
#define KCHUNK  512                      // samples staged in LDS per block
#define KCHUNKS (NBATCH / KCHUNK)        // 16
#define MTILES  (NFEAT / 16)             // 256
#define WPB     8                        // waves per block
#define MGROUPS (MTILES / WPB)           // 32 -> grid = 32*16 = 512 blocks

typedef __attribute__((ext_vector_type(2))) float v2f;
typedef __attribute__((ext_vector_type(8))) float v8f;

// accumulator layout in d_ws (floats):
//  [0] sumX2   [1] dotWY   [2..11] colsum[10]
//  [12..111]  G_H[10][10]  (H_emb^T H_emb)
//  [112..211] G_W[10][10]  (W W^T over features)
#define ACC_SUMX2 0
#define ACC_DOTWY 1
#define ACC_COL   2
#define ACC_GH    12
#define ACC_GW    112
#define ACC_N     256

__device__ __forceinline__ float clamp01(float v) {
    return v < 0.f ? 0.f : (v > 1.f ? 1.f : v);
}

__global__ __launch_bounds__(256) void zero_accum_kernel(float* acc) {
    if (threadIdx.x < ACC_N) acc[threadIdx.x] = 0.f;
}

// colsums + H^T H of the gathered, clamped embedding rows. 32 blocks x 256.
__global__ __launch_bounds__(256) void h_stats_kernel(const int* __restrict__ ids,
                                                      const float* __restrict__ Hw,
                                                      float* __restrict__ acc) {
    __shared__ float s[112];
    const int t = threadIdx.x;
    if (t < 112) s[t] = 0.f;
    __syncthreads();

    const int i = blockIdx.x * 256 + t;          // i < 8192
    float h[NCLS];
    const float* hr = Hw + (long long)ids[i] * NCLS;
    #pragma unroll
    for (int k = 0; k < NCLS; ++k) h[k] = clamp01(hr[k]);
    #pragma unroll
    for (int k = 0; k < NCLS; ++k) atomicAdd(&s[k], h[k]);
    #pragma unroll
    for (int k = 0; k < NCLS; ++k)
        #pragma unroll
        for (int kk = 0; kk < NCLS; ++kk)
            atomicAdd(&s[10 + k * NCLS + kk], h[k] * h[kk]);
    __syncthreads();

    if (t < 10)            atomicAdd(&acc[ACC_COL + t], s[t]);
    else if (t < 110)      atomicAdd(&acc[ACC_GH + (t - 10)], s[t]);
}

// G_W[k][kk] = sum_f W[k,f]*W[kk,f].  16 blocks x 256 (one column f each).
__global__ __launch_bounds__(256) void w_stats_kernel(const float* __restrict__ W,
                                                      float* __restrict__ acc) {
    __shared__ float s[100];
    const int t = threadIdx.x;
    if (t < 100) s[t] = 0.f;
    __syncthreads();

    const int f = blockIdx.x * 256 + t;          // f < 4096
    float w[NCLS];
    #pragma unroll
    for (int k = 0; k < NCLS; ++k) w[k] = W[(long long)k * NFEAT + f];
    #pragma unroll
    for (int k = 0; k < NCLS; ++k)
        #pragma unroll
        for (int kk = 0; kk < NCLS; ++kk)
            atomicAdd(&s[k * NCLS + kk], w[k] * w[kk]);
    __syncthreads();

    if (t < 100) atomicAdd(&acc[ACC_GW + t], s[t]);
}

// Main pass: per block stage 512 clamped H rows (padded to 16 cols) in LDS,
// 8 waves each own a 16-row X tile and run K=512 of f32 WMMA, fusing
// sum(X^2) and the <W, Y_partial> contraction.
__global__ __launch_bounds__(256) void recon_wmma_kernel(const float* __restrict__ X,
                                                         const int* __restrict__ ids,
                                                         const float* __restrict__ Hw,
                                                         const float* __restrict__ W,
                                                         float* __restrict__ acc) {
    __shared__ float Hs[KCHUNK * 16];            // 32 KB
    const int t     = threadIdx.x;
    const int kc    = blockIdx.x & (KCHUNKS - 1);
    const int mg    = blockIdx.x >> 4;           // log2(KCHUNKS) == 4
    const int kbase = kc * KCHUNK;

    // gather clamp(H_weight[ids]) for this K-chunk, zero-pad classes 10..15
    for (int s = t; s < KCHUNK; s += 256) {
        const float* hr = Hw + (long long)ids[kbase + s] * NCLS;
        #pragma unroll
        for (int k = 0; k < NCLS; ++k) Hs[s * 16 + k] = clamp01(hr[k]);
        #pragma unroll
        for (int k = NCLS; k < 16; ++k) Hs[s * 16 + k] = 0.f;
    }
    __syncthreads();

    const int wave = t >> 5;
    const int lane = t & 31;
    const int half = lane >> 4;                  // K-group select (A/B) / M-half (C)
    const int col  = lane & 15;                  // A: M-row,  B/C: N-column
    const int f0   = (mg * WPB + wave) * 16;

    // A fragment: lane owns row f0+col, K pair {4j+2*half, +1} (contiguous 8B)
    const float* xrow = X + (long long)(f0 + col) * NBATCH + kbase + 2 * half;

    v8f  c   = {};
    float xsq = 0.f;

    #pragma unroll 4
    for (int j = 0; j < KCHUNK / 4; ++j) {
        const float2 a2 = *(const float2*)(xrow + 4 * j);
        v2f a; a.x = a2.x; a.y = a2.y;

        const int rbase = (4 * j + 2 * half) * 16 + col;   // conflict-free banks
        v2f b; b.x = Hs[rbase]; b.y = Hs[rbase + 16];

        c = __builtin_amdgcn_wmma_f32_16x16x4_f32(false, a, false, b,
                                                  (short)0, c, false, false);
        xsq += a2.x * a2.x + a2.y * a2.y;        // fused sum(X^2), each elem once
    }

    // C layout: VGPR r, lane -> (M = r + 8*half, N = col).  Contract vs W row.
    float dot = 0.f;
    if (col < NCLS) {
        const float* wrun = W + (long long)col * NFEAT + f0 + 8 * half;
        #pragma unroll
        for (int r = 0; r < 8; ++r) dot += c[r] * wrun[r];
    }

    // wave32 butterfly reduction, then 2 atomics per wave
    #pragma unroll
    for (int ofs = 16; ofs > 0; ofs >>= 1) {
        dot += __shfl_xor(dot, ofs, 32);
        xsq += __shfl_xor(xsq, ofs, 32);
    }
    if (lane == 0) {
        atomicAdd(&acc[ACC_DOTWY], dot);
        atomicAdd(&acc[ACC_SUMX2], xsq);
    }
}

__global__ void finalize_kernel(const float* __restrict__ acc, float* __restrict__ out) {
    const float n = (float)NBATCH;
    float cross = 0.f, sumsqH = 0.f, sumW2 = 0.f, colsq = 0.f;
    for (int k = 0; k < NCLS; ++k) {
        const float cs = acc[ACC_COL + k];
        colsq  += cs * cs;
        sumsqH += acc[ACC_GH + k * NCLS + k];
        sumW2  += acc[ACC_GW + k * NCLS + k];
        for (int kk = 0; kk < NCLS; ++kk)
            cross += acc[ACC_GH + k * NCLS + kk] * acc[ACC_GW + k * NCLS + kk];
    }
    const float recon = (acc[ACC_SUMX2] - 2.f * acc[ACC_DOTWY] + cross) / n;
    const float trace = (colsq - sumsqH) / n;                 // THOU/2 = 0.5
    out[0] = recon + 0.5f * trace + 0.01f * (sumsqH + sumW2) / n;  // (MIUH+MIUW)/2
}

extern "C" void kernel_launch(void* const* d_in, const int* in_sizes, int n_in,
                              void* d_out, int out_size, void* d_ws, size_t ws_size,
                              hipStream_t stream) {
    const float* X   = (const float*)d_in[0];   // [4096, 8192]
    const int*   ids = (const int*)d_in[1];     // [8192]
    const float* W   = (const float*)d_in[2];   // [10, 4096]
    const float* Hw  = (const float*)d_in[3];   // [100000, 10]
    float* acc = (float*)d_ws;
    float* out = (float*)d_out;
    (void)in_sizes; (void)n_in; (void)out_size; (void)ws_size;

    zero_accum_kernel<<<1, 256, 0, stream>>>(acc);
    h_stats_kernel<<<NBATCH / 256, 256, 0, stream>>>(ids, Hw, acc);
    w_stats_kernel<<<NFEAT / 256, 256, 0, stream>>>(W, acc);
    recon_wmma_kernel<<<MGROUPS * KCHUNKS, 256, 0, stream>>>(X, ids, Hw, W, acc);
    finalize_kernel<<<1, 1, 0, stream>>>(acc, out);
}